// ProvidenceLSTM_89309549953090
// MI455X (gfx1250) — compile-verified
//
#include <hip/hip_runtime.h>
#include <cmath>

// ---------------------------------------------------------------------------
// Fused LSTM (T=512,B=2048,I=24,H=20) + exp/softplus heads for gfx1250.
// One wave32 per 16-batch-row tile; per step:
//   gates[16x80] = x_t[16x24] @ W_ih^T  +  h[16x20] @ W_hh^T  + (b_ih+b_hh)
// computed with V_WMMA_F32_16X16X4_F32 (5 N-tiles x (6+5) K-chunks = 55 WMMA).
// Gate tile bounced through LDS for the cross-lane elementwise cell update;
// h kept in LDS, c in registers (10 elems/lane). Single-wave workgroup =>
// same-wave DS ordering, no barriers needed. EXEC is all-ones at every WMMA.
// ---------------------------------------------------------------------------

typedef __attribute__((ext_vector_type(2))) float v2f;
typedef __attribute__((ext_vector_type(8))) float v8f;

#define TT  512
#define BB  2048
#define IN  24
#define HH  20
#define GG  80      // 4*H
#define NT  5       // N tiles of 16 covering 80 gates
#define KCX 6       // IN/4 K-chunks
#define KCH 5       // HH/4 K-chunks

__device__ __forceinline__ float sigf(float x) { return 1.0f / (1.0f + __expf(-x)); }

__device__ __forceinline__ float softplusf(float x) {
    if (x > 20.0f) return x;               // stable: log1p(exp(x)) ~ x
    return log1pf(__expf(x));
}

__global__ void __launch_bounds__(32)
providence_lstm_fused(const float* __restrict__ x,   const int* __restrict__ lens,
                      const float* __restrict__ Wih, const float* __restrict__ Whh,
                      const float* __restrict__ bih, const float* __restrict__ bhh,
                      const float* __restrict__ Wa,  const float* __restrict__ ba,
                      const float* __restrict__ Wb,  const float* __restrict__ bb,
                      float* __restrict__ out)
{
    __shared__ float g_lds[16 * GG];   // gate tile [m][n]
    __shared__ float h_lds[16 * HH];   // hidden state [m][j]
    __shared__ float wa_lds[HH];
    __shared__ float wb_lds[HH];

    const int l   = threadIdx.x;          // 0..31
    const int lm  = l & 15;
    const int off = (l >= 16) ? 2 : 0;    // hi half-wave holds K+2,K+3
    const int b0  = blockIdx.x * 16;      // batch tile origin

    // ---- time-invariant B-matrix fragments (kept in VGPRs) ----------------
    // B[k][n] = W[n][k]; lane holds (B[k0+off][n], B[k0+off+1][n]) = W[n][k..k+1]
    v2f wih[NT][KCX];
#pragma unroll
    for (int nt = 0; nt < NT; ++nt)
#pragma unroll
        for (int kc = 0; kc < KCX; ++kc)
            wih[nt][kc] = *(const v2f*)(Wih + (nt * 16 + lm) * IN + kc * 4 + off);

    v2f whh[NT][KCH];
#pragma unroll
    for (int nt = 0; nt < NT; ++nt)
#pragma unroll
        for (int kc = 0; kc < KCH; ++kc)
            whh[nt][kc] = *(const v2f*)(Whh + (nt * 16 + lm) * HH + kc * 4 + off);

    float bias[NT];
#pragma unroll
    for (int nt = 0; nt < NT; ++nt)
        bias[nt] = bih[nt * 16 + lm] + bhh[nt * 16 + lm];

    if (l < HH) { wa_lds[l] = Wa[l]; wb_lds[l] = Wb[l]; }
    const float bav = ba[0];
    const float bbv = bb[0];
    const int   len = (l < 16) ? lens[b0 + l] : 0;

    // ---- zero initial state ----------------------------------------------
    float creg[10];
#pragma unroll
    for (int r = 0; r < 10; ++r) { creg[r] = 0.0f; h_lds[l + 32 * r] = 0.0f; }

    for (int t = 0; t < TT; ++t) {
        // prefetch next timestep's x tile into cache (global_prefetch_b8)
        {
            const int tn = (t + 1 < TT) ? (t + 1) : t;
            __builtin_prefetch(x + ((size_t)tn * BB + b0 + lm) * IN, 0, 1);
        }

        // A-fragments: x_t rows (16x24) and h rows (16x20)
        v2f xa[KCX];
#pragma unroll
        for (int kc = 0; kc < KCX; ++kc)
            xa[kc] = *(const v2f*)(x + ((size_t)t * BB + b0 + lm) * IN + kc * 4 + off);

        v2f ha[KCH];
#pragma unroll
        for (int kc = 0; kc < KCH; ++kc)
            ha[kc] = *(const v2f*)(h_lds + lm * HH + kc * 4 + off);

        // ---- 5 N-tiles of 16x16 gates: 11 chained f32 WMMAs each ---------
#pragma unroll
        for (int nt = 0; nt < NT; ++nt) {
            v8f acc = {};
#pragma unroll
            for (int kc = 0; kc < KCX; ++kc)
                acc = __builtin_amdgcn_wmma_f32_16x16x4_f32(
                    false, xa[kc], false, wih[nt][kc], (short)0, acc, false, false);
#pragma unroll
            for (int kc = 0; kc < KCH; ++kc)
                acc = __builtin_amdgcn_wmma_f32_16x16x4_f32(
                    false, ha[kc], false, whh[nt][kc], (short)0, acc, false, false);

            // spill gate tile to LDS (+bias); vgpr v -> M = v (+8 hi half)
#pragma unroll
            for (int v = 0; v < 8; ++v) {
                const int m = v + ((l >= 16) ? 8 : 0);
                g_lds[m * GG + nt * 16 + lm] = acc[v] + bias[nt];
            }
        }

        // ---- elementwise LSTM cell: 10 (m,j) state elements per lane -----
#pragma unroll
        for (int r = 0; r < 10; ++r) {
            const int s = l + 32 * r;       // 0..319
            const int m = s / HH;
            const int j = s - m * HH;
            const float* gm = g_lds + m * GG;
            const float iv = gm[j];
            const float fv = gm[HH + j];
            const float gv = gm[2 * HH + j];
            const float ov = gm[3 * HH + j];
            const float c  = sigf(fv) * creg[r] + sigf(iv) * tanhf(gv);
            creg[r] = c;
            h_lds[m * HH + j] = sigf(ov) * tanhf(c);   // raw (unmasked) h
        }

        // ---- alpha/beta heads with length masking (lanes 0..15, m = lane)
        if (l < 16) {
            float da = 0.0f, db = 0.0f;
#pragma unroll
            for (int j = 0; j < HH; ++j) {
                const float hv = h_lds[l * HH + j];
                da += hv * wa_lds[j];
                db += hv * wb_lds[j];
            }
            const bool  valid = (t < len);          // mask zeroes hs, not bias
            const float aarg  = valid ? (da + bav) : bav;
            const float barg  = valid ? (db + bbv) : bbv;
            const size_t idx  = (size_t)t * BB + b0 + l;
            out[idx]                    = __expf(aarg);       // alpha
            out[(size_t)TT * BB + idx]  = softplusf(barg);    // beta
        }
    }
}

extern "C" void kernel_launch(void* const* d_in, const int* in_sizes, int n_in,
                              void* d_out, int out_size, void* d_ws, size_t ws_size,
                              hipStream_t stream) {
    const float* x   = (const float*)d_in[0];
    const int*   len = (const int*)  d_in[1];
    const float* Wih = (const float*)d_in[2];
    const float* Whh = (const float*)d_in[3];
    const float* bih = (const float*)d_in[4];
    const float* bhh = (const float*)d_in[5];
    const float* Wa  = (const float*)d_in[6];
    const float* ba  = (const float*)d_in[7];
    const float* Wb  = (const float*)d_in[8];
    const float* bb  = (const float*)d_in[9];
    float* out = (float*)d_out;

    dim3 grid(BB / 16);   // 128 independent batch-tile waves
    dim3 block(32);       // one wave32 per workgroup
    hipLaunchKernelGGL(providence_lstm_fused, grid, block, 0, stream,
                       x, len, Wih, Whh, bih, bhh, Wa, ba, Wb, bb, out);

    (void)in_sizes; (void)n_in; (void)out_size; (void)d_ws; (void)ws_size;
}